// PointTransformerLayer_25220047962574
// MI455X (gfx1250) — compile-verified
//
#include <hip/hip_runtime.h>
#include <hip/hip_bf16.h>

typedef __attribute__((ext_vector_type(2))) float v2f;
typedef __attribute__((ext_vector_type(8))) float v8f;

#define NPTS   8192
#define NTOT   16384   // B*N
#define KNN    16
#define CIN    256
#define CMID   128
#define COUT   256

// ---------------------------------------------------------------------------
// Kernel 1: C[n,Cout] = A[n,256] @ W[256,Cout] + bias, via V_WMMA_F32_16X16X4_F32
// Block: 256 threads = 8 waves. Block tile: 128 rows x 64 cols.
// Each wave owns a 16x64 strip = 4 independent 16x16 accumulators:
//   - breaks the WMMA RAW accumulator chain 4-wide (XDL stays fed)
//   - one A fragment feeds 4 WMMAs (4x less A-side LDS traffic per FLOP)
// K staged through LDS in 64-wide chunks; next A chunk prefetched.
// ---------------------------------------------------------------------------
__global__ __launch_bounds__(256) void gemm_proj_kernel(
    const float* __restrict__ A, const float* __restrict__ W,
    const float* __restrict__ bias, float* __restrict__ C, int Cout)
{
    __shared__ float As[128][68];   // +4 pad: conflict-free column reads
    __shared__ float Ws[64][80];    // +16 pad: half-wave bank sets disjoint

    const int lane  = threadIdx.x & 31;
    const int wave  = threadIdx.x >> 5;          // 0..7
    const int mrow  = lane & 15;                 // 0..15
    const int khalf = (lane >> 4) << 1;          // 0 or 2
    const int col0  = blockIdx.y * 64;

    v8f acc0 = {}, acc1 = {}, acc2 = {}, acc3 = {};

    for (int k0 = 0; k0 < CIN; k0 += 64) {
        // stage A tile (128x64), coalesced: 32 floats/thread
        for (int i = threadIdx.x; i < 128 * 64; i += 256) {
            int r = i >> 6, c = i & 63;
            As[r][c] = A[(size_t)(blockIdx.x * 128 + r) * CIN + k0 + c];
        }
        // stage W tile (64x64): 16 floats/thread
        for (int i = threadIdx.x; i < 64 * 64; i += 256) {
            int r = i >> 6, c = i & 63;
            Ws[r][c] = W[(size_t)(k0 + r) * Cout + col0 + c];
        }
        // prefetch next A chunk into cache while this chunk computes
        if (k0 + 64 < CIN) {
            int r = threadIdx.x >> 1, c = (threadIdx.x & 1) << 5;
            __builtin_prefetch(&A[(size_t)(blockIdx.x * 128 + r) * CIN + k0 + 64 + c], 0, 2);
        }
        __syncthreads();

        #pragma unroll
        for (int kk = 0; kk < 64; kk += 4) {
            v2f a, b0, b1, b2, b3;
            // A frag (16x4 f32): lane holds A[M=lane%16][K=2*(lane/16)+v]
            a.x = As[wave * 16 + mrow][kk + khalf + 0];
            a.y = As[wave * 16 + mrow][kk + khalf + 1];
            // B frags (4x16 f32): lane holds B[K=2*(lane/16)+v][N=lane%16]
            b0.x = Ws[kk + khalf + 0][mrow];      b0.y = Ws[kk + khalf + 1][mrow];
            b1.x = Ws[kk + khalf + 0][mrow + 16]; b1.y = Ws[kk + khalf + 1][mrow + 16];
            b2.x = Ws[kk + khalf + 0][mrow + 32]; b2.y = Ws[kk + khalf + 1][mrow + 32];
            b3.x = Ws[kk + khalf + 0][mrow + 48]; b3.y = Ws[kk + khalf + 1][mrow + 48];
            acc0 = __builtin_amdgcn_wmma_f32_16x16x4_f32(false, a, false, b0, (short)0, acc0, false, false);
            acc1 = __builtin_amdgcn_wmma_f32_16x16x4_f32(false, a, false, b1, (short)0, acc1, false, false);
            acc2 = __builtin_amdgcn_wmma_f32_16x16x4_f32(false, a, false, b2, (short)0, acc2, false, false);
            acc3 = __builtin_amdgcn_wmma_f32_16x16x4_f32(false, a, false, b3, (short)0, acc3, false, false);
        }
        __syncthreads();
    }

    // D layout: lane, vgpr v -> D[M = v + 8*(lane/16)][N = lane%16]
    const float bc0 = bias[col0 + mrow];
    const float bc1 = bias[col0 + 16 + mrow];
    const float bc2 = bias[col0 + 32 + mrow];
    const float bc3 = bias[col0 + 48 + mrow];
    const int   rbase = blockIdx.x * 128 + wave * 16 + (lane >> 4) * 8;
    #pragma unroll
    for (int v = 0; v < 8; ++v) {
        size_t row = (size_t)(rbase + v) * Cout + col0;
        C[row + mrow]      = acc0[v] + bc0;
        C[row + 16 + mrow] = acc1[v] + bc1;
        C[row + 32 + mrow] = acc2[v] + bc2;
        C[row + 48 + mrow] = acc3[v] + bc3;
    }
}

// ---------------------------------------------------------------------------
// Kernel 2: kNN (top-16 smallest squared distances) over xyz, per batch.
// One thread per query point; candidates tiled through LDS (2048/chunk).
// Sorted top-16 held in registers (constant indices only -> no spill).
// ---------------------------------------------------------------------------
__global__ __launch_bounds__(256) void knn_kernel(
    const float* __restrict__ xyz, int* __restrict__ knn_out)
{
    const int CH = 2048;
    __shared__ float xs[CH], ys[CH], zs[CH];

    const int b = blockIdx.y;
    const float* base = xyz + (size_t)b * NPTS * 3;
    const int q = blockIdx.x * 256 + threadIdx.x;

    const float qx = base[q * 3 + 0];
    const float qy = base[q * 3 + 1];
    const float qz = base[q * 3 + 2];

    float best[KNN];
    int   bidx[KNN];
    #pragma unroll
    for (int k = 0; k < KNN; ++k) { best[k] = 3.4e38f; bidx[k] = 0; }

    for (int c0 = 0; c0 < NPTS; c0 += CH) {
        __syncthreads();
        for (int i = threadIdx.x; i < CH; i += 256) {
            xs[i] = base[(c0 + i) * 3 + 0];
            ys[i] = base[(c0 + i) * 3 + 1];
            zs[i] = base[(c0 + i) * 3 + 2];
        }
        __syncthreads();

        #pragma unroll 4
        for (int j = 0; j < CH; ++j) {
            float dx = xs[j] - qx, dy = ys[j] - qy, dz = zs[j] - qz;
            float d  = dx * dx + dy * dy + dz * dz;
            if (d < best[KNN - 1]) {            // rare path
                best[KNN - 1] = d;
                bidx[KNN - 1] = c0 + j;
                #pragma unroll
                for (int k = KNN - 1; k > 0; --k) {
                    if (best[k] < best[k - 1]) {
                        float tf = best[k]; best[k] = best[k - 1]; best[k - 1] = tf;
                        int   ti = bidx[k]; bidx[k] = bidx[k - 1]; bidx[k - 1] = ti;
                    }
                }
            }
        }
    }

    int* o = knn_out + ((size_t)b * NPTS + q) * KNN;
    #pragma unroll
    for (int k = 0; k < KNN; ++k) o[k] = bidx[k];
}

// ---------------------------------------------------------------------------
// Kernel 3: fused point-transformer attention. One block (256 thr) per point.
// ---------------------------------------------------------------------------
__global__ __launch_bounds__(256) void attn_kernel(
    const float* __restrict__ xyz,
    const float* __restrict__ qf,   // [NTOT,128]
    const float* __restrict__ kf,   // [NTOT,128]
    const float* __restrict__ vf,   // [NTOT,256]
    const int*   __restrict__ knn,  // [NTOT,16]
    const float* __restrict__ p1_w, const float* __restrict__ p1_b,
    const float* __restrict__ pbn_g, const float* __restrict__ pbn_b,
    const float* __restrict__ pbn_m, const float* __restrict__ pbn_v,
    const float* __restrict__ p2_w, const float* __restrict__ p2_b,
    const float* __restrict__ bn1_g, const float* __restrict__ bn1_b,
    const float* __restrict__ bn1_m, const float* __restrict__ bn1_v,
    const float* __restrict__ w1_w, const float* __restrict__ w1_b,
    const float* __restrict__ bn2_g, const float* __restrict__ bn2_b,
    const float* __restrict__ bn2_m, const float* __restrict__ bn2_v,
    const float* __restrict__ w2_w, const float* __restrict__ w2_b,
    float* __restrict__ out)
{
    __shared__ int   nidx[KNN];
    __shared__ float rel[KNN][3];
    __shared__ float u3[KNN][3];
    __shared__ float prf[KNN][260];   // p_r, padded (bank-safe)
    __shared__ float wpre[KNN][132];  // relu(bn1(w)), padded
    __shared__ float hk[KNN][20];     // relu(bn2(w@w1))
    __shared__ float lg[KNN][36];     // logits
    __shared__ float wsm[KNN][36];    // softmax weights

    const int p = blockIdx.x;         // global point id
    const int b = p >> 13;            // /8192
    const int t = threadIdx.x;        // channel id 0..255

    if (t < KNN) nidx[t] = knn[(size_t)p * KNN + t];
    __syncthreads();

    // relative positions rel[k][d] = xyz[nbr] - xyz[self]
    if (t < KNN * 3) {
        int k = t / 3, d = t - 3 * k;
        int j = nidx[k];
        rel[k][d] = xyz[((size_t)b * NPTS + j) * 3 + d] - xyz[(size_t)p * 3 + d];
    }
    __syncthreads();

    // linear_p stage 1: Linear(3,3) -> BN(3) -> ReLU
    if (t < KNN * 3) {
        int k = t / 3, d = t - 3 * k;
        float s = p1_b[d];
        #pragma unroll
        for (int e = 0; e < 3; ++e) s += rel[k][e] * p1_w[e * 3 + d];
        s = (s - pbn_m[d]) * (pbn_g[d] * rsqrtf(pbn_v[d] + 1e-5f)) + pbn_b[d];
        u3[k][d] = fmaxf(s, 0.f);
    }
    __syncthreads();

    // linear_p stage 2: p_r[k][c] = u3[k] . p2_w[:,c] + p2_b[c]   (c = t)
    {
        float w0 = p2_w[0 * COUT + t], w1 = p2_w[1 * COUT + t];
        float w2 = p2_w[2 * COUT + t], bb = p2_b[t];
        #pragma unroll
        for (int k = 0; k < KNN; ++k)
            prf[k][t] = u3[k][0] * w0 + u3[k][1] * w1 + u3[k][2] * w2 + bb;
    }
    __syncthreads();

    // w = x_k - x_q + fold(p_r);  relu(bn1(w))  (channels 0..127)
    if (t < CMID) {
        float xq = qf[(size_t)p * CMID + t];
        float sc = bn1_g[t] * rsqrtf(bn1_v[t] + 1e-5f);
        float mm = bn1_m[t], bb = bn1_b[t];
        #pragma unroll
        for (int k = 0; k < KNN; ++k) {
            float xk = kf[((size_t)b * NPTS + nidx[k]) * CMID + t];
            float w  = xk - xq + prf[k][t] + prf[k][t + CMID];
            w = (w - mm) * sc + bb;
            wpre[k][t] = fmaxf(w, 0.f);
        }
    }
    __syncthreads();

    // w @ w1_w (128->16), bn2, relu : thread t -> (k = t/16, j = t%16)
    {
        int k = t >> 4, j = t & 15;
        float s = w1_b[j];
        for (int c = 0; c < CMID; ++c) s += wpre[k][c] * w1_w[c * 16 + j];
        s = (s - bn2_m[j]) * (bn2_g[j] * rsqrtf(bn2_v[j] + 1e-5f)) + bn2_b[j];
        hk[k][j] = fmaxf(s, 0.f);
    }
    __syncthreads();

    // w @ w2_w (16->32): 512 logits, 2 per thread
    #pragma unroll
    for (int r = 0; r < 2; ++r) {
        int e = t + 256 * r;
        int k = e >> 5, jj = e & 31;
        float s = w2_b[jj];
        #pragma unroll
        for (int j = 0; j < 16; ++j) s += hk[k][j] * w2_w[j * 32 + jj];
        lg[k][jj] = s;
    }
    __syncthreads();

    // softmax over the K neighbors, per channel jj (32 channels)
    if (t < 32) {
        float mx = -3.4e38f;
        #pragma unroll
        for (int k = 0; k < KNN; ++k) mx = fmaxf(mx, lg[k][t]);
        float ex[KNN];
        float den = 0.f;
        #pragma unroll
        for (int k = 0; k < KNN; ++k) { ex[k] = __expf(lg[k][t] - mx); den += ex[k]; }
        float inv = 1.f / den;
        #pragma unroll
        for (int k = 0; k < KNN; ++k) wsm[k][t] = ex[k] * inv;
    }
    __syncthreads();

    // out[c] = sum_k (x_v[k][c] + p_r[k][c]) * wsm[k][c%32]
    {
        int jj = t & 31;
        float acc = 0.f;
        #pragma unroll
        for (int k = 0; k < KNN; ++k) {
            float xv = vf[((size_t)b * NPTS + nidx[k]) * COUT + t];
            acc += (xv + prf[k][t]) * wsm[k][jj];
        }
        out[(size_t)p * COUT + t] = acc;
    }
}

// ---------------------------------------------------------------------------
extern "C" void kernel_launch(void* const* d_in, const int* in_sizes, int n_in,
                              void* d_out, int out_size, void* d_ws, size_t ws_size,
                              hipStream_t stream)
{
    (void)in_sizes; (void)n_in; (void)out_size; (void)ws_size;

    const float* xyz    = (const float*)d_in[0];
    const float* points = (const float*)d_in[1];
    const float* Wq = (const float*)d_in[2];  const float* bq = (const float*)d_in[3];
    const float* Wk = (const float*)d_in[4];  const float* bk = (const float*)d_in[5];
    const float* Wv = (const float*)d_in[6];  const float* bv = (const float*)d_in[7];
    const float* p1_w  = (const float*)d_in[8];   const float* p1_b  = (const float*)d_in[9];
    const float* pbn_g = (const float*)d_in[10];  const float* pbn_b = (const float*)d_in[11];
    const float* pbn_m = (const float*)d_in[12];  const float* pbn_v = (const float*)d_in[13];
    const float* p2_w  = (const float*)d_in[14];  const float* p2_b  = (const float*)d_in[15];
    const float* bn1_g = (const float*)d_in[16];  const float* bn1_b = (const float*)d_in[17];
    const float* bn1_m = (const float*)d_in[18];  const float* bn1_v = (const float*)d_in[19];
    const float* w1_w  = (const float*)d_in[20];  const float* w1_b  = (const float*)d_in[21];
    const float* bn2_g = (const float*)d_in[22];  const float* bn2_b = (const float*)d_in[23];
    const float* bn2_m = (const float*)d_in[24];  const float* bn2_v = (const float*)d_in[25];
    const float* w2_w  = (const float*)d_in[26];  const float* w2_b  = (const float*)d_in[27];

    // workspace layout
    float* q   = (float*)d_ws;                      // [16384,128]  8 MB
    float* kfe = q   + (size_t)NTOT * CMID;         // [16384,128]  8 MB
    float* vfe = kfe + (size_t)NTOT * CMID;         // [16384,256] 16 MB
    int*   knn = (int*)(vfe + (size_t)NTOT * COUT); // [16384,16]   1 MB

    dim3 blk(256);

    // Q/K/V projections (WMMA f32 16x16x4, 128x64 block tiles)
    gemm_proj_kernel<<<dim3(NTOT / 128, CMID / 64), blk, 0, stream>>>(points, Wq, bq, q,   CMID);
    gemm_proj_kernel<<<dim3(NTOT / 128, CMID / 64), blk, 0, stream>>>(points, Wk, bk, kfe, CMID);
    gemm_proj_kernel<<<dim3(NTOT / 128, COUT / 64), blk, 0, stream>>>(points, Wv, bv, vfe, COUT);

    // kNN (self-grouping, top-16 by squared distance)
    knn_kernel<<<dim3(NPTS / 256, 2), blk, 0, stream>>>(xyz, knn);

    // fused vector attention
    attn_kernel<<<dim3(NTOT), blk, 0, stream>>>(
        xyz, q, kfe, vfe, knn,
        p1_w, p1_b, pbn_g, pbn_b, pbn_m, pbn_v, p2_w, p2_b,
        bn1_g, bn1_b, bn1_m, bn1_v, w1_w, w1_b,
        bn2_g, bn2_b, bn2_m, bn2_v, w2_w, w2_b,
        (float*)d_out);
}